// GuidedSelfAttention_77876347011143
// MI455X (gfx1250) — compile-verified
//
#include <hip/hip_runtime.h>
#include <hip/hip_bf16.h>

// ---------------------------------------------------------------------------
// GuidedSelfAttention, exploiting the tiled structure of the reference:
//   S columns repeat with period T  ->  only S_small = neigh @ node^T [1024x64]
//   softmax(tiled) = softmax_small / N
//   output = softmax_small @ V_avg,  V_avg = mean_i neigh[b,i,:,:]
//   W[b,i,t] = (1/N) * colsum_q softmax_small  (broadcast over i)
// Memory-bound (~300MB @ 23.3TB/s); all matmuls via V_WMMA_F32_16X16X4_F32.
// ---------------------------------------------------------------------------

typedef __attribute__((ext_vector_type(2))) float v2f;
typedef __attribute__((ext_vector_type(8))) float v8f;

#define Bb 32
#define Nn 16
#define Tt 64
#define Hh 256
#define NT 1024          // Nn*Tt
#define QT 64            // queries per workgroup (4 waves x 16 rows)
#define NS 260           // LDS row stride for node/vavg (260 % 64 == 4 -> conflict-free frags)
#define PS 68            // LDS row stride for P tile   (68  % 64 == 4)

// ---------------- Kernel 1: V_avg[b,t,h] = mean_i neigh[b,i,t,h] -------------
__global__ __launch_bounds__(256) void vavg_kernel(const float* __restrict__ neigh,
                                                   float* __restrict__ vavg) {
    int idx = blockIdx.x * 256 + threadIdx.x;      // over B*T*H/4 = 131072
    int bt  = idx >> 6;                            // H/4 = 64 vec4 per (b,t)
    int h4  = idx & 63;
    int b   = bt >> 6;                             // T = 64
    int t   = bt & 63;
    const float4* src = (const float4*)neigh + ((size_t)(b * Nn * Tt + t) * (Hh / 4)) + h4;
    float4 s = {0.f, 0.f, 0.f, 0.f};
#pragma unroll
    for (int i = 0; i < Nn; ++i) {
        float4 v = src[(size_t)i * Tt * (Hh / 4)];
        s.x += v.x; s.y += v.y; s.z += v.z; s.w += v.w;
    }
    const float r = 1.0f / (float)Nn;
    float4 o = {s.x * r, s.y * r, s.z * r, s.w * r};
    ((float4*)vavg)[idx] = o;
}

// ---------------- Kernel 2: fused S -> softmax -> (P@Vavg, colsums) ----------
__global__ __launch_bounds__(128) void attn_kernel(const float* __restrict__ node,   // [B,T,H]
                                                   const float* __restrict__ neigh,  // [B,NT,H]
                                                   const float* __restrict__ vavg,   // [B,T,H]
                                                   const long long* __restrict__ nn, // [B]
                                                   float* __restrict__ out,          // [B,NT,H]
                                                   float* __restrict__ wpart)        // [B*64,T]
{
    __shared__ float node_lds[Tt * NS];      // 66,560 B
    __shared__ float vavg_lds[Tt * NS];      // 66,560 B
    __shared__ float p_lds[4][16 * PS];      // 17,408 B  (per-wave 16x64 P tiles)

    const int b    = blockIdx.x >> 4;        // 16 WGs per batch (1024/QT)
    const int qblk = blockIdx.x & 15;

    // stage node[b] and V_avg[b] into LDS (fully coalesced float4)
    const float4* nsrc = (const float4*)(node + (size_t)b * Tt * Hh);
    const float4* vsrc = (const float4*)(vavg + (size_t)b * Tt * Hh);
    for (int i = threadIdx.x; i < Tt * Hh / 4; i += 128) {
        int t = i >> 6;
        int c = (i & 63) << 2;
        *(float4*)&node_lds[t * NS + c] = nsrc[i];
        *(float4*)&vavg_lds[t * NS + c] = vsrc[i];
    }
    __syncthreads();

    const int wave = threadIdx.x >> 5;
    const int lane = threadIdx.x & 31;
    const int half = lane >> 4;              // lane-half selects K pair per ISA layout
    const int lh   = lane & 15;
    const float inv_scale = rsqrtf((float)nn[b]);

    const int   qrow = qblk * QT + wave * 16;                  // this wave's 16 query rows
    const float* A   = neigh + ((size_t)b * NT + qrow) * Hh;   // A-matrix rows (row-major)

    const v8f vzero = {0.f, 0.f, 0.f, 0.f, 0.f, 0.f, 0.f, 0.f};

    // ---- GEMM1: S_tile[16x64] = A[16x256] x node^T[256x64], K in steps of 4
    v8f acc[4];
#pragma unroll
    for (int nt = 0; nt < 4; ++nt) acc[nt] = vzero;

#pragma unroll 4
    for (int k0 = 0; k0 < Hh; k0 += 4) {
        // A frag: lanes 0-15 -> rows, K={k0,k0+1}; lanes 16-31 -> same rows, K={k0+2,k0+3}
        v2f a = *(const v2f*)(A + (size_t)lh * Hh + (k0 + 2 * half));
#pragma unroll
        for (int nt = 0; nt < 4; ++nt) {
            // B frag: B[k][n] = node[n][k]; vgpr j, half h -> K = k0+2h+j, N = lh
            v2f bb = *(const v2f*)&node_lds[(nt * 16 + lh) * NS + k0 + 2 * half];
            acc[nt] = __builtin_amdgcn_wmma_f32_16x16x4_f32(false, a, false, bb,
                                                            (short)0, acc[nt], false, false);
        }
    }

    // ---- scale + spill S tile to LDS (C layout: vgpr r, half h -> row r+8h, col=lane)
    float* P = p_lds[wave];
#pragma unroll
    for (int nt = 0; nt < 4; ++nt)
#pragma unroll
        for (int r = 0; r < 8; ++r)
            P[(r + 8 * half) * PS + nt * 16 + lh] = acc[nt][r] * inv_scale;
    __syncthreads();

    // ---- softmax over the 64 distinct logits (lanes 0-15 each own one row)
    if (lane < 16) {
        float* row = P + lane * PS;
        float m = -3.402823466e38f;
#pragma unroll 8
        for (int c = 0; c < Tt; ++c) m = fmaxf(m, row[c]);
        float s = 0.f;
#pragma unroll 8
        for (int c = 0; c < Tt; ++c) { float e = __expf(row[c] - m); row[c] = e; s += e; }
        float is = 1.0f / s;
#pragma unroll 8
        for (int c = 0; c < Tt; ++c) row[c] *= is;
    }
    __syncthreads();

    // ---- per-wave column sums of normalized P -> workspace (deterministic, no atomics)
    {
        const int gw = blockIdx.x * 4 + wave;
#pragma unroll
        for (int cc = 0; cc < 2; ++cc) {
            int c = lane + cc * 32;
            float s = 0.f;
#pragma unroll
            for (int r = 0; r < 16; ++r) s += P[r * PS + c];
            wpart[(size_t)gw * Tt + c] = s;
        }
    }

    // ---- GEMM2: out_tile[16x256] = P[16x64] x Vavg[64x256]
    float* orow = out + ((size_t)b * NT + qrow) * Hh;
    for (int nh = 0; nh < 16; ++nh) {
        v8f o = vzero;
#pragma unroll
        for (int k0 = 0; k0 < Tt; k0 += 4) {
            v2f a = *(const v2f*)&P[lh * PS + k0 + 2 * half];
            v2f bb;
            bb.x = vavg_lds[(k0 + 2 * half) * NS + nh * 16 + lh];
            bb.y = vavg_lds[(k0 + 2 * half + 1) * NS + nh * 16 + lh];
            o = __builtin_amdgcn_wmma_f32_16x16x4_f32(false, a, false, bb,
                                                      (short)0, o, false, false);
        }
#pragma unroll
        for (int r = 0; r < 8; ++r)
            orow[(size_t)(r + 8 * half) * Hh + nh * 16 + lh] = o[r];
    }
}

// ---------------- Kernel 3: W[b,i,t] = (1/N) * sum_w wpart[b*64+w][t] --------
__global__ __launch_bounds__(256) void wout_kernel(const float* __restrict__ wpart,
                                                   float* __restrict__ wout) {
    int idx = blockIdx.x * 256 + threadIdx.x;     // over B*N*T = 32768
    if (idx >= Bb * Nn * Tt) return;
    int t = idx & 63;
    int b = idx >> 10;                            // N*T = 1024
    float s = 0.f;
#pragma unroll 8
    for (int w = 0; w < 64; ++w) s += wpart[(size_t)(b * 64 + w) * Tt + t];
    wout[idx] = s * (1.0f / (float)Nn);
}

extern "C" void kernel_launch(void* const* d_in, const int* in_sizes, int n_in,
                              void* d_out, int out_size, void* d_ws, size_t ws_size,
                              hipStream_t stream) {
    const float*     node  = (const float*)d_in[0];       // [B,T,H]
    const float*     neigh = (const float*)d_in[1];       // [B,N,T,H]
    const long long* nn    = (const long long*)d_in[2];   // [B] int64

    float* out  = (float*)d_out;                          // [B,NT,H] then W [B,N,T]
    float* wout = out + (size_t)Bb * NT * Hh;

    float* vavg  = (float*)d_ws;                          // B*T*H floats = 2 MB
    float* wpart = vavg + (size_t)Bb * Tt * Hh;           // B*64*T floats = 512 KB

    vavg_kernel<<<(Bb * Tt * Hh / 4) / 256, 256, 0, stream>>>(neigh, vavg);
    attn_kernel<<<Bb * (NT / QT), 128, 0, stream>>>(node, neigh, vavg, nn, out, wpart);
    wout_kernel<<<(Bb * Nn * Tt) / 256, 256, 0, stream>>>(wpart, wout);
}